// KANLayer_74517682586063
// MI455X (gfx1250) — compile-verified
//
#include <hip/hip_runtime.h>

typedef __attribute__((ext_vector_type(2))) float v2f;
typedef __attribute__((ext_vector_type(8))) float v8f;

constexpr int D = 256;   // input dim (K)
constexpr int C = 3;     // components
constexpr int G = 6;     // grid size
constexpr int O = 256;   // output dim
constexpr int THREADS = 256;          // 8 waves (wave32)
constexpr int ROWS_PER_BLOCK = 128;   // 16 rows per wave

__global__ __launch_bounds__(THREADS)
void kan_fused_kernel(const float* __restrict__ x,              // [N, D]
                      const float* __restrict__ projections,    // [D, C]
                      const float* __restrict__ control_points, // [C, G, O]
                      const float* __restrict__ comp_weights,   // [C]
                      float* __restrict__ out,                  // [N, O]
                      int nrows)
{
    __shared__ float s_B[D * 16];                 // projections padded [D][16], 16 KB
    __shared__ float s_cp[C * G * O];             // control points, 18 KB
    __shared__ float s_proj[ROWS_PER_BLOCK * C];  // clipped proj per row, 1.5 KB

    const int tid = threadIdx.x;

    // ---- Stage B (zero-padded 3 -> 16 columns) and control points into LDS ----
    for (int i = tid; i < D * 16; i += THREADS) {
        const int k = i >> 4;
        const int n = i & 15;
        s_B[i] = (n < C) ? projections[k * C + n] : 0.0f;
    }
    for (int i = tid; i < C * G * O; i += THREADS)
        s_cp[i] = control_points[i];
    __syncthreads();

    // ---- Phase 1: projection GEMM tile via V_WMMA_F32_16X16X4_F32 ----
    // Wave owns 16 rows. A(16x4 f32): lane<16 -> K={0,1}, lane>=16 -> K={2,3}.
    // B(4x16 f32): same half-lane K offset pattern; column = lane & 15.
    const int wave = tid >> 5;
    const int lane = tid & 31;
    const int half = lane >> 4;   // 0: low half lanes, 1: high half lanes
    const int n    = lane & 15;
    const int rowBase = blockIdx.x * ROWS_PER_BLOCK + wave * 16;

    int arow = rowBase + n;                  // A-matrix row for this lane
    if (arow >= nrows) arow = nrows - 1;     // clamp so EXEC stays all-ones
    const float* xr = x + (size_t)arow * D;

    v8f acc = {};
    #pragma unroll 4
    for (int k = 0; k < D / 4; ++k) {
        const int kk = k * 4 + half * 2;
        v2f a, b;
        a.x = xr[kk];                        // global_load_b64 (8B aligned)
        a.y = xr[kk + 1];
        b.x = s_B[kk * 16 + n];
        b.y = s_B[(kk + 1) * 16 + n];
        // 8 args: (neg_a, A, neg_b, B, c_mod, C, reuse_a, reuse_b)
        acc = __builtin_amdgcn_wmma_f32_16x16x4_f32(
            false, a, false, b, (short)0, acc, false, false);
    }

    // D-tile layout: VGPR j, lane -> M = j + 8*half, N = lane&15.
    // Lanes with n < 3 hold component n for this wave's 16 rows.
    if (n < C) {
        #pragma unroll
        for (int j = 0; j < 8; ++j) {
            float p = acc[j];
            p = fminf(fmaxf(p, -0.99f), 0.99f);   // reference clip
            s_proj[(wave * 16 + half * 8 + j) * C + n] = p;
        }
    }
    __syncthreads();

    // ---- Phase 2: lerp over control points + weighted sum + tanh ----
    float w[C];
    #pragma unroll
    for (int c = 0; c < C; ++c) w[c] = comp_weights[c];

    const int o = tid;  // one output column per thread -> coalesced stores
    for (int r = 0; r < ROWS_PER_BLOCK; ++r) {
        const int grow = blockIdx.x * ROWS_PER_BLOCK + r;
        if (grow >= nrows) break;
        float sum = 0.0f;
        #pragma unroll
        for (int c = 0; c < C; ++c) {
            const float p = s_proj[r * C + c];          // LDS broadcast
            // grid = linspace(-1,1,6): spacing exactly 0.4
            int idx = (int)((p + 1.0f) * 2.5f);         // (G-1)/2 = 2.5
            idx = idx < 0 ? 0 : (idx > G - 2 ? G - 2 : idx);
            const float g0 = -1.0f + 0.4f * (float)idx;
            const float t  = (p - g0) * 2.5f;           // / 0.4
            const float p0 = s_cp[(c * G + idx)     * O + o];
            const float p1 = s_cp[(c * G + idx + 1) * O + o];
            sum += w[c] * fmaf(t, p1 - p0, p0);
        }
        out[(size_t)grow * O + o] = tanhf(sum);
    }
}

extern "C" void kernel_launch(void* const* d_in, const int* in_sizes, int n_in,
                              void* d_out, int out_size, void* d_ws, size_t ws_size,
                              hipStream_t stream) {
    const float* x  = (const float*)d_in[0];  // [32, 2048, 256] f32
    const float* P  = (const float*)d_in[1];  // [256, 3] f32
    const float* CP = (const float*)d_in[2];  // [3, 6, 256] f32
    const float* W  = (const float*)d_in[3];  // [3] f32
    float* out = (float*)d_out;               // [32, 2048, 256] f32

    const int nrows  = in_sizes[0] / D;
    const int blocks = (nrows + ROWS_PER_BLOCK - 1) / ROWS_PER_BLOCK;

    kan_fused_kernel<<<blocks, THREADS, 0, stream>>>(x, P, CP, W, out, nrows);
}